// GResBlock_25975962206483
// MI455X (gfx1250) — compile-verified
//
#include <hip/hip_runtime.h>

#define DIM    128
#define NODES  50000
#define EDGES  800000
#define MTILES (NODES / 16)   // 3125
#define NTILES (DIM / 16)     // 8

typedef __attribute__((ext_vector_type(2))) float v2f;
typedef __attribute__((ext_vector_type(8))) float v8f;

// ---------------------------------------------------------------- utilities
__global__ void zero_f32(float* __restrict__ p, int n) {
    int i = blockIdx.x * blockDim.x + threadIdx.x;
    if (i < n) p[i] = 0.0f;
}

// deg[dst] += 1 per edge (hardware FP32 atomic, stays in L2)
__global__ void deg_count(const int* __restrict__ dst, float* __restrict__ deg, int E) {
    int i = blockIdx.x * blockDim.x + threadIdx.x;
    if (i < E) unsafeAtomicAdd(&deg[dst[i]], 1.0f);
}

// dinv[i] = rsqrt(deg[i] + 1)  (self-loop included), in place
__global__ void finalize_dinv(float* __restrict__ dinv, int n) {
    int i = blockIdx.x * blockDim.x + threadIdx.x;
    if (i < n) dinv[i] = rsqrtf(dinv[i] + 1.0f);
}

// ---------------------------------------------------------------- WMMA GEMM
// H[M,128] = X[M,128] @ W[128,128], f32, one 16x16 tile per wave,
// K accumulated in 32 steps of V_WMMA_F32_16X16X4_F32.
__global__ __launch_bounds__(128) void wmma_gemm128(const float* __restrict__ X,
                                                    const float* __restrict__ W,
                                                    float* __restrict__ H) {
    const int wavesPerBlock = blockDim.x >> 5;
    const int tile = blockIdx.x * wavesPerBlock + (threadIdx.x >> 5);
    const int mt = tile >> 3;          // tile / NTILES
    const int nt = tile & 7;           // tile % NTILES
    if (mt >= MTILES) return;

    const int lane = threadIdx.x & 31;
    const int half = lane >> 4;        // 0: K=k,k+1   1: K=k+2,k+3
    const int idx  = lane & 15;        // A row / B col within tile

    // A: lane holds row (mt*16+idx), K pair selected by half
    const float* arow = X + (size_t)(mt * 16 + idx) * DIM + half * 2;
    // B: lane holds col (nt*16+idx), rows (k+half*2, k+half*2+1)
    const float* bcol = W + (size_t)(half * 2) * DIM + nt * 16 + idx;

    v8f acc = {};
    #pragma unroll 4
    for (int k = 0; k < DIM; k += 4) {
        v2f a; a.x = arow[k];            a.y = arow[k + 1];
        v2f b; b.x = bcol[k * DIM];      b.y = bcol[k * DIM + DIM];
        acc = __builtin_amdgcn_wmma_f32_16x16x4_f32(
            /*neg_a=*/false, a, /*neg_b=*/false, b,
            /*c_mod=*/(short)0, acc, /*reuse_a=*/false, /*reuse_b=*/false);
    }

    // C/D layout: VGPR v -> M = v + half*8, N = idx
    float* drow = H + nt * 16 + idx;
    #pragma unroll
    for (int v = 0; v < 8; ++v)
        drow[(size_t)(mt * 16 + v + half * 8) * DIM] = acc[v];
}

// ---------------------------------------------------------------- aggregation
// agg[i,:] = H[i,:] * dinv[i]^2   (self-loop term). One float4 per thread.
__global__ void self_loop_init(const float* __restrict__ H,
                               const float* __restrict__ dinv,
                               float* __restrict__ agg) {
    int i4 = blockIdx.x * blockDim.x + threadIdx.x;   // float4 index
    if (i4 >= NODES * (DIM / 4)) return;
    int node = i4 >> 5;                               // 32 float4 per row
    float s = dinv[node]; s = s * s;
    float4 v = ((const float4*)H)[i4];
    v.x *= s; v.y *= s; v.z *= s; v.w *= s;
    ((float4*)agg)[i4] = v;
}

// agg[dst,:] += H[src,:] * dinv[src]*dinv[dst].  One wave per edge:
// 32 lanes x float4 = full 128-dim row, coalesced load + FP32 atomics (L2).
__global__ void edge_scatter(const float* __restrict__ H,
                             const int*   __restrict__ src,
                             const int*   __restrict__ dst,
                             const float* __restrict__ dinv,
                             float*       __restrict__ agg, int E) {
    int gw   = (blockIdx.x * blockDim.x + threadIdx.x) >> 5;
    int nw   = (gridDim.x * blockDim.x) >> 5;
    int lane = threadIdx.x & 31;
    for (int e = gw; e < E; e += nw) {
        int s = src[e], d = dst[e];
        float nrm = dinv[s] * dinv[d];
        float4 v = ((const float4*)(H + (size_t)s * DIM))[lane];
        float* ap = agg + (size_t)d * DIM + lane * 4;
        unsafeAtomicAdd(ap + 0, v.x * nrm);
        unsafeAtomicAdd(ap + 1, v.y * nrm);
        unsafeAtomicAdd(ap + 2, v.z * nrm);
        unsafeAtomicAdd(ap + 3, v.w * nrm);
    }
}

// agg = relu(agg + b), in place
__global__ void bias_relu(float* __restrict__ agg, const float* __restrict__ b) {
    int i4 = blockIdx.x * blockDim.x + threadIdx.x;
    if (i4 >= NODES * (DIM / 4)) return;
    float4 v  = ((float4*)agg)[i4];
    float4 bb = ((const float4*)b)[i4 & 31];
    v.x = fmaxf(v.x + bb.x, 0.0f);
    v.y = fmaxf(v.y + bb.y, 0.0f);
    v.z = fmaxf(v.z + bb.z, 0.0f);
    v.w = fmaxf(v.w + bb.w, 0.0f);
    ((float4*)agg)[i4] = v;
}

// out = relu(agg + b + x)   (residual)
__global__ void final_out(const float* __restrict__ agg, const float* __restrict__ b,
                          const float* __restrict__ x,   float* __restrict__ out) {
    int i4 = blockIdx.x * blockDim.x + threadIdx.x;
    if (i4 >= NODES * (DIM / 4)) return;
    float4 v  = ((const float4*)agg)[i4];
    float4 bb = ((const float4*)b)[i4 & 31];
    float4 xx = ((const float4*)x)[i4];
    v.x = fmaxf(v.x + bb.x + xx.x, 0.0f);
    v.y = fmaxf(v.y + bb.y + xx.y, 0.0f);
    v.z = fmaxf(v.z + bb.z + xx.z, 0.0f);
    v.w = fmaxf(v.w + bb.w + xx.w, 0.0f);
    ((float4*)out)[i4] = v;
}

// ---------------------------------------------------------------- launcher
extern "C" void kernel_launch(void* const* d_in, const int* in_sizes, int n_in,
                              void* d_out, int out_size, void* d_ws, size_t ws_size,
                              hipStream_t stream) {
    const float* x  = (const float*)d_in[0];
    const int*   ei = (const int*)  d_in[1];
    const float* W1 = (const float*)d_in[2];
    const float* b1 = (const float*)d_in[3];
    const float* W2 = (const float*)d_in[4];
    const float* b2 = (const float*)d_in[5];
    float* out = (float*)d_out;

    float* ws   = (float*)d_ws;
    float* dinv = ws;                         // 50000 floats (padded to 51200)
    float* h    = ws + 51200;                 // 6.4M floats
    float* agg  = h + (size_t)NODES * DIM;    // 6.4M floats

    const int* src = ei;                      // edge_index[0]
    const int* dst = ei + EDGES;              // edge_index[1]

    const int elem4Blocks = (NODES * (DIM / 4) + 255) / 256;   // 6250
    const int gemmBlocks  = (MTILES * NTILES) / 4;             // 6250 (4 waves/blk)

    // normalization
    zero_f32     <<<(NODES + 255) / 256, 256, 0, stream>>>(dinv, NODES);
    deg_count    <<<(EDGES + 255) / 256, 256, 0, stream>>>(dst, dinv, EDGES);
    finalize_dinv<<<(NODES + 255) / 256, 256, 0, stream>>>(dinv, NODES);

    // ---- layer 1 ----
    wmma_gemm128  <<<gemmBlocks, 128, 0, stream>>>(x, W1, h);
    self_loop_init<<<elem4Blocks, 256, 0, stream>>>(h, dinv, agg);
    edge_scatter  <<<4096, 256, 0, stream>>>(h, src, dst, dinv, agg, EDGES);
    bias_relu     <<<elem4Blocks, 256, 0, stream>>>(agg, b1);      // agg -> out1

    // ---- layer 2 ----
    wmma_gemm128  <<<gemmBlocks, 128, 0, stream>>>(agg, W2, h);    // out1 consumed
    self_loop_init<<<elem4Blocks, 256, 0, stream>>>(h, dinv, agg); // agg reused
    edge_scatter  <<<4096, 256, 0, stream>>>(h, src, dst, dinv, agg, EDGES);
    final_out     <<<elem4Blocks, 256, 0, stream>>>(agg, b2, x, out);
}